// GraphLSTMCell_33921651704146
// MI455X (gfx1250) — compile-verified
//
#include <hip/hip_runtime.h>
#include <hip/hip_bf16.h>
#include <stdint.h>

#define BATCH 16384
#define DIM   256
#define UNITS 256
#define TT    32
#define MM    64
#define KTOT  512          // DIM + UNITS (concat K)
#define NZ    1024         // 4*UNITS

#define KC    8            // K-chunk for the main GEMM
#define BPITCH 1032        // 1024 + 2 pad dwords per 256 (TDM pad) -> conflict-free
#define APITCH 12          // 8 + 4 pad dwords per 8 (TDM pad)      -> conflict-free

typedef __attribute__((ext_vector_type(2))) float v2f;
typedef __attribute__((ext_vector_type(8))) float v8f;
typedef __attribute__((ext_vector_type(4))) unsigned int vu4;
typedef __attribute__((ext_vector_type(8))) int vi8;
typedef __attribute__((ext_vector_type(4))) int vi4;

#if defined(__has_builtin)
#if __has_builtin(__builtin_amdgcn_tensor_load_to_lds) && __has_builtin(__builtin_amdgcn_s_wait_tensorcnt)
#define USE_TDM 1
#endif
#endif
#ifndef USE_TDM
#define USE_TDM 0
#endif

__device__ __forceinline__ float hsig(float x) {
  return fminf(fmaxf(0.2f * x + 0.5f, 0.0f), 1.0f);
}

__device__ __forceinline__ v8f wmma_f32_16x16x4(v2f a, v2f b, v8f c) {
#if defined(__has_builtin) && __has_builtin(__builtin_amdgcn_wmma_f32_16x16x4_f32)
  // 8 args: (neg_a, A, neg_b, B, c_mod, C, reuse_a, reuse_b)
  return __builtin_amdgcn_wmma_f32_16x16x4_f32(false, a, false, b, (short)0, c,
                                               false, false);
#else
  c[0] += a[0] * b[0] + a[1] * b[1];  // placeholder fallback (keeps compiling)
  return c;
#endif
}

// ---------------------------------------------------------------------------
// Kernel 1: column sums of h_tm1 / h_tm2 over the batch  -> hs1, hs2 (256 each)
// ---------------------------------------------------------------------------
__global__ __launch_bounds__(256) void colsum_kernel(
    const float* __restrict__ h1, const float* __restrict__ h2,
    float* __restrict__ hs1, float* __restrict__ hs2) {
  const int col = threadIdx.x;            // 0..255
  const int r0 = blockIdx.x * 128;        // 128 rows per block
  float s1 = 0.f, s2 = 0.f;
  for (int r = 0; r < 128; ++r) {
    s1 += h1[(size_t)(r0 + r) * UNITS + col];
    s2 += h2[(size_t)(r0 + r) * UNITS + col];
  }
  atomicAdd(&hs1[col], s1);
  atomicAdd(&hs2[col], s2);
}

// ---------------------------------------------------------------------------
// Kernel 2: pack concat weights  Wz = [W;U] (512x1024), Wf = [W_f;Un_f] (512x256)
// ---------------------------------------------------------------------------
__global__ __launch_bounds__(256) void pack_weights_kernel(
    const float* __restrict__ W, const float* __restrict__ U,
    const float* __restrict__ Un, float* __restrict__ Wz,
    float* __restrict__ Wf) {
  const int idx = blockIdx.x * 256 + threadIdx.x;
  if (idx < KTOT * NZ) {
    const int k = idx >> 10, n = idx & (NZ - 1);
    Wz[idx] = (k < DIM) ? W[k * NZ + n] : U[(k - DIM) * NZ + n];
  } else {
    const int j = idx - KTOT * NZ;        // 0 .. 512*256-1
    const int k = j >> 8, n = j & 255;
    Wf[j] = (k < DIM) ? W[k * NZ + UNITS + n] : Un[(k - DIM) * NZ + UNITS + n];
  }
}

// ---------------------------------------------------------------------------
// Kernel 3: per-row neighbor scalars a[b]=has*cnt_past/denom, cf[b]=has*cnt_fut/denom
// ---------------------------------------------------------------------------
__global__ __launch_bounds__(256) void neighbor_scalars_kernel(
    const int* __restrict__ ng, const int* __restrict__ mp,
    const int* __restrict__ rv, const int* __restrict__ time_p,
    float* __restrict__ a_out, float* __restrict__ cf_out) {
  const int b = blockIdx.x * blockDim.x + threadIdx.x;
  const int t = *time_p;
  const int prev = rv[(size_t)b * TT + t];
  const int* row = ng + ((size_t)b * TT + prev) * MM;
  const int* mrow = mp + (size_t)b * MM;
  float num = 0.f, past = 0.f;
  for (int m = 0; m < MM; ++m) {
    const float v = (float)row[m];
    num += v;
    if (mrow[m] < t) past += v;
  }
  const float has = (num > 0.f) ? 1.f : 0.f;
  const float denom = (num > 0.f) ? num : 1.f;
  a_out[b] = has * past / denom;
  cf_out[b] = has * (num - past) / denom;
}

// ---------------------------------------------------------------------------
// Kernel 4: v1 = hs1 @ Un, v2 = hs2 @ Un   (1024 each)
// ---------------------------------------------------------------------------
__global__ __launch_bounds__(256) void vecmat_kernel(
    const float* __restrict__ hs1, const float* __restrict__ hs2,
    const float* __restrict__ Un, float* __restrict__ v1,
    float* __restrict__ v2) {
  const int n = blockIdx.x * 256 + threadIdx.x;  // 0..1023
  float s1 = 0.f, s2 = 0.f;
  for (int k = 0; k < UNITS; ++k) {
    const float u = Un[k * NZ + n];
    s1 += hs1[k] * u;
    s2 += hs2[k] * u;
  }
  v1[n] = s1;
  v2[n] = s2;
}

// ---------------------------------------------------------------------------
// Kernel 5: f_avg GEMM  Zf = [inputs|h_tm1] @ Wf + bias_f ; m1/m2 column sums
//   tile: 32 rows x 256 cols per block; 8 waves = 2 rowgroups x 4 colgroups
// ---------------------------------------------------------------------------
__global__ __launch_bounds__(256) void gemm_favg_kernel(
    const float* __restrict__ inputs, const float* __restrict__ h_tm1,
    const float* __restrict__ Wf, const float* __restrict__ bias,
    const float* __restrict__ c_tm1, const float* __restrict__ c_tm2,
    float* __restrict__ m1, float* __restrict__ m2) {
  __shared__ float ldsA[32][36];
  const int tid = threadIdx.x;
  const int wave = tid >> 5, lane = tid & 31;
  const int rg = wave >> 2;   // 0..1
  const int cg = wave & 3;    // 0..3
  const int lc = lane & 15, half = lane >> 4;
  const int row0 = blockIdx.x * 32;

  v8f acc[4] = {};

  for (int chunk = 0; chunk < KTOT / 32; ++chunk) {
    const int k0 = chunk * 32;
    const float* Asrc = (k0 < DIM) ? (inputs + (size_t)row0 * DIM + k0)
                                   : (h_tm1 + (size_t)row0 * UNITS + (k0 - DIM));
    __syncthreads();
    {
      const int r = tid >> 3;          // 0..31
      const int c4 = (tid & 7) * 4;    // 0..28
      const float* src = Asrc + (size_t)r * 256 + c4;
      ldsA[r][c4 + 0] = src[0];
      ldsA[r][c4 + 1] = src[1];
      ldsA[r][c4 + 2] = src[2];
      ldsA[r][c4 + 3] = src[3];
    }
    __syncthreads();

    for (int kk = 0; kk < 32; kk += 4) {
      const int kb = kk + half * 2;
      v2f af = {ldsA[rg * 16 + lc][kb], ldsA[rg * 16 + lc][kb + 1]};
      const float* Bp = Wf + (size_t)(k0 + kb) * UNITS;
#pragma unroll
      for (int q = 0; q < 4; ++q) {
        const int col = cg * 64 + q * 16 + lc;
        v2f bf = {Bp[col], Bp[UNITS + col]};
        acc[q] = wmma_f32_16x16x4(af, bf, acc[q]);
      }
    }
  }

#pragma unroll
  for (int q = 0; q < 4; ++q) {
    const int n = cg * 64 + q * 16 + lc;
    const float bb = bias[UNITS + n];
    float s1 = 0.f, s2 = 0.f;
#pragma unroll
    for (int v = 0; v < 8; ++v) {
      const int b = row0 + rg * 16 + v + half * 8;
      const float favg = hsig(acc[q][v] + bb);
      s1 += favg * c_tm1[(size_t)b * UNITS + n];
      s2 += favg * c_tm2[(size_t)b * UNITS + n];
    }
    s1 += __shfl_xor(s1, 16, 32);
    s2 += __shfl_xor(s2, 16, 32);
    if (half == 0) {
      atomicAdd(&m1[n], s1);
      atomicAdd(&m2[n], s2);
    }
  }
}

// ---------------------------------------------------------------------------
// Kernel 6: main gate GEMM  Z = [inputs|h_tm1] @ [W;U] + bias, fused epilogue.
//   tile 32 rows x 1024 cols; wave (rg,cg) owns cols {g*256 + cg*64 + q*16}
//   for all 4 gates g -> gate fusion is wave-local.
//   Per K-chunk (KC=8) the Tensor Data Mover stages BOTH operands into LDS:
//     A tile 32x8  (pad 4dw/8dw  -> pitch 12,  bank-conflict-free frag reads)
//     B slab 8x1024 (pad 2dw/256dw -> pitch 1032, half-waves hit disjoint banks)
//   B slab is a contiguous region of the packed Wz, shared by all 8 waves
//   (removes the duplicated per-wave global_load_b32 stream of the previous rev).
// ---------------------------------------------------------------------------
__global__ __launch_bounds__(256) void gemm_gates_kernel(
    const float* __restrict__ inputs, const float* __restrict__ h_tm1,
    const float* __restrict__ Wz, const float* __restrict__ bias,
    const float* __restrict__ v1, const float* __restrict__ v2,
    const float* __restrict__ m1, const float* __restrict__ m2,
    const float* __restrict__ a_arr, const float* __restrict__ cf_arr,
    const float* __restrict__ c_tm1, float* __restrict__ out) {
  __shared__ float ldsA[32][APITCH];      // 1.5 KB
  __shared__ float ldsB[KC * BPITCH];     // 33 KB
  const int tid = threadIdx.x;
  const int wave = tid >> 5, lane = tid & 31;
  const int rg = wave >> 2;   // 0..1 (16-row groups)
  const int cg = wave & 3;    // 0..3 (64-col group within each gate)
  const int lc = lane & 15, half = lane >> 4;
  const int row0 = blockIdx.x * 32;

  v8f acc[16] = {};

  for (int chunk = 0; chunk < KTOT / KC; ++chunk) {
    const int k0 = chunk * KC;
    const float* Asrc = (k0 < DIM) ? (inputs + (size_t)row0 * DIM + k0)
                                   : (h_tm1 + (size_t)row0 * UNITS + (k0 - DIM));
    const float* Bsrc = Wz + (size_t)k0 * NZ;   // contiguous 8x1024 slab
    __syncthreads();
#if USE_TDM
    if (tid < 32) {
      const unsigned long long gaA = (unsigned long long)(uintptr_t)Asrc;
      const unsigned ldsA_off = (unsigned)(uintptr_t)(&ldsA[0][0]);
      // ---- A tile: 32 rows x 8 cols of a row-stride-256 f32 matrix ----
      vu4 a0 = {1u, ldsA_off, (unsigned)gaA,
                (unsigned)(((gaA >> 32) & 0x01FFFFFFull) | (2u << 30))};
      // data_size=4B | pad_en | pad_interval=8dw(code2) | pad_amount=4dw(code3)
      vi8 a1 = {(int)((2u << 16) | (1u << 20) | (2u << 22) | (3u << 25)),
                (int)(256u << 16),     // tensor_dim0[15:0]=256
                (int)(16384u << 16),   // tensor_dim1[15:0]=16384
                (int)((unsigned)KC << 16),  // tile_dim0=8
                32,                    // tile_dim1=32
                256,                   // tensor_dim0_stride=256
                0, 0};
      // ---- B slab: 8 rows x 1024 cols, contiguous in packed Wz ----
      const unsigned long long gaB = (unsigned long long)(uintptr_t)Bsrc;
      const unsigned ldsB_off = (unsigned)(uintptr_t)(&ldsB[0]);
      vu4 b0 = {1u, ldsB_off, (unsigned)gaB,
                (unsigned)(((gaB >> 32) & 0x01FFFFFFull) | (2u << 30))};
      // data_size=4B | pad_en | pad_interval=256dw(code7) | pad_amount=2dw(code1)
      vi8 b1 = {(int)((2u << 16) | (1u << 20) | (7u << 22) | (1u << 25)),
                (int)(1024u << 16),    // tensor_dim0[15:0]=1024
                (int)(512u << 16),     // tensor_dim1[15:0]=512
                (int)(1024u << 16),    // tile_dim0=1024
                KC,                    // tile_dim1=8
                1024,                  // tensor_dim0_stride=1024
                0, 0};
      vi4 zz = {0, 0, 0, 0};
      vi8 z8 = {0, 0, 0, 0, 0, 0, 0, 0};
      __builtin_amdgcn_tensor_load_to_lds(a0, a1, zz, zz, z8, 0);
      __builtin_amdgcn_tensor_load_to_lds(b0, b1, zz, zz, z8, 0);
      __builtin_amdgcn_s_wait_tensorcnt(0);
    }
#else
    {
      // A tile: 32x8, one float per thread, padded pitch
      const int r = tid >> 3, c = tid & 7;
      ldsA[r][c] = Asrc[(size_t)r * 256 + c];
      // B slab: 8x1024, 32 floats per thread, padded pitch 1032 (+2 per 256)
      const int br = tid >> 5;              // 0..7
      for (int j = 0; j < 8; ++j) {
        const int c4 = (tid & 31) * 4 + j * 128;
        const float* src = Bsrc + (size_t)br * NZ + c4;
        const int di = br * BPITCH + c4 + 2 * (c4 >> 8);
        ldsB[di + 0] = src[0];
        ldsB[di + 1] = src[1];
        ldsB[di + 2] = src[2];
        ldsB[di + 3] = src[3];
      }
    }
#endif
    __syncthreads();

#pragma unroll
    for (int kk = 0; kk < KC; kk += 4) {
      const int kb = kk + half * 2;
      v2f af = {ldsA[rg * 16 + lc][kb], ldsA[rg * 16 + lc][kb + 1]};
      const int bbase = kb * BPITCH + cg * 64 + lc;
#pragma unroll
      for (int g = 0; g < 4; ++g) {
#pragma unroll
        for (int q = 0; q < 4; ++q) {
          const int bi = bbase + g * 258 + q * 16;   // 258 = 256 cols + 2 pad
          v2f bf = {ldsB[bi], ldsB[bi + BPITCH]};
          acc[g * 4 + q] = wmma_f32_16x16x4(af, bf, acc[g * 4 + q]);
        }
      }
    }
  }

  // Fused gate epilogue (wave-local: this wave holds i/f/c/o for its columns)
  float* out_h = out;
  float* out_m = out + (size_t)BATCH * UNITS;
#pragma unroll
  for (int q = 0; q < 4; ++q) {
    const int n = cg * 64 + q * 16 + lc;
    const float b_i = bias[n];
    const float b_f = bias[UNITS + n];
    const float b_c = bias[2 * UNITS + n];
    const float b_o = bias[3 * UNITS + n];
    const float V1i = v1[n], V2i = v2[n];
    const float V1c = v1[2 * UNITS + n], V2c = v2[2 * UNITS + n];
    const float V1o = v1[3 * UNITS + n], V2o = v2[3 * UNITS + n];
    const float M1 = m1[n], M2 = m2[n];
#pragma unroll
    for (int v = 0; v < 8; ++v) {
      const int b = row0 + rg * 16 + v + half * 8;
      const float aa = a_arr[b];
      const float cc = cf_arr[b];
      const float zi = acc[q][v] + b_i;
      const float zf = acc[4 + q][v] + b_f;
      const float zc = acc[8 + q][v] + b_c;
      const float zo = acc[12 + q][v] + b_o;
      const float ig = hsig(zi + aa * V1i + cc * V2i);
      const float fg = hsig(zf);
      const float cgate = tanhf(zc + aa * V1c + cc * V2c);
      const float og = hsig(zo + aa * V1o + cc * V2o);
      const float mem =
          aa * M1 + cc * M2 + fg * c_tm1[(size_t)b * UNITS + n] + ig * cgate;
      out_h[(size_t)b * UNITS + n] = og * tanhf(mem);
      out_m[(size_t)b * UNITS + n] = mem;
    }
  }
}

// ---------------------------------------------------------------------------
// Host-side launcher
// ---------------------------------------------------------------------------
extern "C" void kernel_launch(void* const* d_in, const int* in_sizes, int n_in,
                              void* d_out, int out_size, void* d_ws,
                              size_t ws_size, hipStream_t stream) {
  (void)in_sizes; (void)n_in; (void)out_size; (void)ws_size;

  const float* inputs = (const float*)d_in[0];
  const float* h_tm1 = (const float*)d_in[1];
  const float* c_tm1 = (const float*)d_in[2];
  const float* h_tm2 = (const float*)d_in[3];
  const float* c_tm2 = (const float*)d_in[4];
  const int* neighbors = (const int*)d_in[5];
  const int* mapping = (const int*)d_in[6];
  const int* revmap = (const int*)d_in[7];
  const int* time_p = (const int*)d_in[8];
  const float* W = (const float*)d_in[9];
  const float* U = (const float*)d_in[10];
  const float* Un = (const float*)d_in[11];
  const float* bias = (const float*)d_in[12];
  float* out = (float*)d_out;
  float* ws = (float*)d_ws;

  // workspace layout (floats)
  float* hs1 = ws;                         // 256
  float* hs2 = ws + 256;                   // 256
  float* m1 = ws + 512;                    // 256
  float* m2 = ws + 768;                    // 256
  float* v1 = ws + 1024;                   // 1024
  float* v2 = ws + 2048;                   // 1024
  float* a_arr = ws + 3072;                // 16384
  float* cf_arr = ws + 3072 + BATCH;       // 16384
  float* Wf = ws + 3072 + 2 * BATCH;       // 512*256
  float* Wz = Wf + KTOT * UNITS;           // 512*1024

  // zero the accumulators (hs1, hs2, m1, m2)
  (void)hipMemsetAsync(d_ws, 0, 1024 * sizeof(float), stream);

  colsum_kernel<<<BATCH / 128, 256, 0, stream>>>(h_tm1, h_tm2, hs1, hs2);
  pack_weights_kernel<<<(KTOT * NZ + KTOT * UNITS) / 256, 256, 0, stream>>>(
      W, U, Un, Wz, Wf);
  neighbor_scalars_kernel<<<BATCH / 256, 256, 0, stream>>>(
      neighbors, mapping, revmap, time_p, a_arr, cf_arr);
  vecmat_kernel<<<4, 256, 0, stream>>>(hs1, hs2, Un, v1, v2);
  gemm_favg_kernel<<<BATCH / 32, 256, 0, stream>>>(inputs, h_tm1, Wf, bias,
                                                   c_tm1, c_tm2, m1, m2);
  gemm_gates_kernel<<<BATCH / 32, 256, 0, stream>>>(
      inputs, h_tm1, Wz, bias, v1, v2, m1, m2, a_arr, cf_arr, c_tm1, out);
}